// Model_2430951489964
// MI455X (gfx1250) — compile-verified
//
#include <hip/hip_runtime.h>
#include <hip/hip_bf16.h>
#include <math.h>

// ---------------------------------------------------------------------------
// Exact separable NUFFT (type-1/type-2 NUDFT) + Pipe density compensation on
// CDNA5 fp32 matrix cores (V_WMMA_F32_16X16X4_F32, wave32).
//
// Both transform directions are cast as complex GEMMs with K = 16384 (traj
// samples) or K = 256 (image columns).  The 64 MiB Ax/Ay DFT tables live in
// the 192 MB L2, so the GEMMs are compute-bound; register blocking (16x32 /
// 32x32 per wave) raises FLOP/byte so the WMMA pipe, not L2, is the limiter.
// ---------------------------------------------------------------------------

typedef float v2f __attribute__((ext_vector_type(2)));
typedef float v8f __attribute__((ext_vector_type(8)));

#define IMG_H 256
#define IMG_W 256
#define TWO_PI_F 6.283185307179586f
#define PIPE_ITERS 10

// D = A(16x4) * B(4x16) + C(16x16), all fp32.
__device__ __forceinline__ v8f wmma4(v2f a, v2f b, v8f c) {
  return __builtin_amdgcn_wmma_f32_16x16x4_f32(false, a, false, b, (short)0, c,
                                               false, false);
}

__device__ __forceinline__ float wave_sum32(float v) {
#pragma unroll
  for (int off = 16; off > 0; off >>= 1) v += __shfl_xor(v, off, 32);
  return v;
}

// --------------------------- table construction ----------------------------
// Ax[m,i] = exp(-i*2pi*kx_m*(i-128));  Ay[m,j] likewise.  [M][256] planar.
__global__ void build_tables(const float* __restrict__ traj,
                             float* __restrict__ AxRe, float* __restrict__ AxIm,
                             float* __restrict__ AyRe, float* __restrict__ AyIm,
                             int M) {
  int idx = blockIdx.x * blockDim.x + threadIdx.x;
  if (idx >= M * 256) return;
  int m = idx >> 8;
  int i = idx & 255;
  float g = (float)(i - 128);
  float tx = TWO_PI_F * traj[2 * m + 0] * g;
  float ty = TWO_PI_F * traj[2 * m + 1] * g;
  float sx, cx, sy, cy;
  __sincosf(tx, &sx, &cx);
  __sincosf(ty, &sy, &cy);
  AxRe[idx] = cx; AxIm[idx] = -sx;
  AyRe[idx] = cy; AyIm[idx] = -sy;
}

__global__ void init_state(float* __restrict__ w, float* __restrict__ accum,
                           int M) {
  int t = blockIdx.x * blockDim.x + threadIdx.x;
  if (t < M) w[t] = 1.0f;
  if (t == 0) *accum = 0.0f;
}

// --------------------------- adjoint (gridding) ----------------------------
// out[i,j] = sum_m g[m] * conj(Ax[m,i]) * conj(Ay[m,j])
// One block per 16x32 output block; one A-tile feeds two B-tiles (8 WMMA per
// A fetch).  8 waves split K = M; partial tiles reduced through LDS.
// Conjugates are folded into the WMMA sign pattern:
//   accRe += aRe*ayRe + aIm*ayIm ;  accIm += aIm*ayRe - aRe*ayIm
__global__ void __launch_bounds__(256) grid_adjoint_wmma(
    const float* __restrict__ gRe, const float* __restrict__ gIm, int gComplex,
    const float* __restrict__ AxRe, const float* __restrict__ AxIm,
    const float* __restrict__ AyRe, const float* __restrict__ AyIm,
    float* __restrict__ outRe, float* __restrict__ outIm, int M) {
  __shared__ float ldsRe[8 * 512];
  __shared__ float ldsIm[8 * 512];

  const int lane = threadIdx.x & 31;
  const int wave = threadIdx.x >> 5;
  const int i0 = (blockIdx.x >> 3) << 4;   // 16 i-tiles
  const int j0 = (blockIdx.x & 7) << 5;    // 8 j-superblocks of 32
  const int half = lane >> 4;              // lanes 16-31 carry K={2,3}
  const int il = lane & 15;
  const int kchunk = M >> 3;
  const int mBeg = wave * kchunk;
  const int mEnd = mBeg + kchunk;
  const v2f* gRe2 = (const v2f*)gRe;       // g[m],g[m+1] pairs -> b64 loads
  const v2f* gIm2 = (const v2f*)gIm;

  v8f acc0Re = {}, acc0Im = {}, acc1Re = {}, acc1Im = {};
  for (int m0 = mBeg; m0 < mEnd; m0 += 4) {
    const int mA = m0 + 2 * half;
    // A operand: a[il, k] = g[m] * conj(Ax[m, i0+il])
    const float ar0 = AxRe[mA * 256 + i0 + il];
    const float ai0 = AxIm[mA * 256 + i0 + il];
    const float ar1 = AxRe[(mA + 1) * 256 + i0 + il];
    const float ai1 = AxIm[(mA + 1) * 256 + i0 + il];
    const v2f gr = gRe2[mA >> 1];
    v2f gi = {};
    if (gComplex) gi = gIm2[mA >> 1];
    v2f aRe, aIm;
    aRe.x = gr.x * ar0 + gi.x * ai0;  aIm.x = gi.x * ar0 - gr.x * ai0;
    aRe.y = gr.y * ar1 + gi.y * ai1;  aIm.y = gi.y * ar1 - gr.y * ai1;
    const v2f aReN = -aRe;
    // B operands: raw Ay for two adjacent j-tiles (conj folded into signs)
    v2f b0Re, b0Im, b1Re, b1Im;
    b0Re.x = AyRe[mA * 256 + j0 + il];
    b0Im.x = AyIm[mA * 256 + j0 + il];
    b0Re.y = AyRe[(mA + 1) * 256 + j0 + il];
    b0Im.y = AyIm[(mA + 1) * 256 + j0 + il];
    b1Re.x = AyRe[mA * 256 + j0 + 16 + il];
    b1Im.x = AyIm[mA * 256 + j0 + 16 + il];
    b1Re.y = AyRe[(mA + 1) * 256 + j0 + 16 + il];
    b1Im.y = AyIm[(mA + 1) * 256 + j0 + 16 + il];
    // 8 WMMAs per K-step (A reused across both j-tiles)
    acc0Re = wmma4(aRe, b0Re, acc0Re);  acc0Re = wmma4(aIm, b0Im, acc0Re);
    acc0Im = wmma4(aIm, b0Re, acc0Im);  acc0Im = wmma4(aReN, b0Im, acc0Im);
    acc1Re = wmma4(aRe, b1Re, acc1Re);  acc1Re = wmma4(aIm, b1Im, acc1Re);
    acc1Im = wmma4(aIm, b1Re, acc1Im);  acc1Im = wmma4(aReN, b1Im, acc1Im);
  }

#pragma unroll
  for (int v = 0; v < 8; ++v) {
    ldsRe[wave * 512 +       v * 32 + lane] = acc0Re[v];
    ldsRe[wave * 512 + 256 + v * 32 + lane] = acc1Re[v];
    ldsIm[wave * 512 +       v * 32 + lane] = acc0Im[v];
    ldsIm[wave * 512 + 256 + v * 32 + lane] = acc1Im[v];
  }
  __syncthreads();
  const int f = threadIdx.x;               // flat (vgpr, lane)
  const int v = f >> 5, l = f & 31;
  const int row = v + 8 * (l >> 4);        // C/D layout: M = vgpr + 8*(lane>=16)
  const int col = l & 15;
#pragma unroll
  for (int t = 0; t < 2; ++t) {
    float sRe = 0.0f, sIm = 0.0f;
#pragma unroll
    for (int wv = 0; wv < 8; ++wv) {
      sRe += ldsRe[wv * 512 + t * 256 + f];
      sIm += ldsIm[wv * 512 + t * 256 + f];
    }
    outRe[(i0 + row) * IMG_W + j0 + t * 16 + col] = sRe;
    outIm[(i0 + row) * IMG_W + j0 + t * 16 + col] = sIm;
  }
}

// ---------------------- forward stage 1 (de-gridding) ----------------------
// V[m,i] = sum_j Ay[m,j] * img[i,j]   (GEMM: M x 256, K = 256)
// One wave per 32x32 output block (2 m-tiles x 2 i-tiles): 16 WMMAs per
// K-step; the (j, j+1) K-pair is contiguous, so operands load as b64.
__global__ void __launch_bounds__(256) degrid_stage1_wmma(
    const float* __restrict__ imgRe, const float* __restrict__ imgIm,
    const float* __restrict__ AyRe, const float* __restrict__ AyIm,
    float* __restrict__ VRe, float* __restrict__ VIm, int M) {
  const int lane = threadIdx.x & 31;
  const int wave = threadIdx.x >> 5;
  const int gid = blockIdx.x * 8 + wave;
  const int nTileI2 = IMG_H >> 5;          // 8 superblocks of 32
  const int m0 = (gid / nTileI2) << 5;
  const int i0 = (gid % nTileI2) << 5;
  if (m0 >= M) return;
  const int half = lane >> 4;
  const int il = lane & 15;
  const v2f* AyRe2 = (const v2f*)AyRe;     // [row*128 + j/2]
  const v2f* AyIm2 = (const v2f*)AyIm;
  const v2f* imRe2 = (const v2f*)imgRe;
  const v2f* imIm2 = (const v2f*)imgIm;
  const int rA0 = (m0 + il) * 128;         // A-tile rows (v2f index base)
  const int rA1 = (m0 + 16 + il) * 128;
  const int rB0 = (i0 + il) * 128;         // B-tile rows
  const int rB1 = (i0 + 16 + il) * 128;

  v8f a00Re = {}, a00Im = {}, a01Re = {}, a01Im = {};
  v8f a10Re = {}, a10Im = {}, a11Re = {}, a11Im = {};
  for (int jk = 0; jk < IMG_W; jk += 4) {
    const int jH = (jk >> 1) + half;       // v2f index of K-pair
    // A: Ay rows m0+il / m0+16+il (forward: no conjugate)
    const v2f A0Re = AyRe2[rA0 + jH], A0Im = AyIm2[rA0 + jH];
    const v2f A1Re = AyRe2[rA1 + jH], A1Im = AyIm2[rA1 + jH];
    const v2f A0ImN = -A0Im, A1ImN = -A1Im;
    // B: img rows i0+il / i0+16+il
    const v2f B0Re = imRe2[rB0 + jH], B0Im = imIm2[rB0 + jH];
    const v2f B1Re = imRe2[rB1 + jH], B1Im = imIm2[rB1 + jH];
    // 16 WMMAs per K-step
    a00Re = wmma4(A0Re, B0Re, a00Re);  a00Re = wmma4(A0ImN, B0Im, a00Re);
    a00Im = wmma4(A0Re, B0Im, a00Im);  a00Im = wmma4(A0Im, B0Re, a00Im);
    a01Re = wmma4(A0Re, B1Re, a01Re);  a01Re = wmma4(A0ImN, B1Im, a01Re);
    a01Im = wmma4(A0Re, B1Im, a01Im);  a01Im = wmma4(A0Im, B1Re, a01Im);
    a10Re = wmma4(A1Re, B0Re, a10Re);  a10Re = wmma4(A1ImN, B0Im, a10Re);
    a10Im = wmma4(A1Re, B0Im, a10Im);  a10Im = wmma4(A1Im, B0Re, a10Im);
    a11Re = wmma4(A1Re, B1Re, a11Re);  a11Re = wmma4(A1ImN, B1Im, a11Re);
    a11Im = wmma4(A1Re, B1Im, a11Im);  a11Im = wmma4(A1Im, B1Re, a11Im);
  }
  const int row = 8 * (lane >> 4);
  const int col = lane & 15;
#pragma unroll
  for (int v = 0; v < 8; ++v) {
    const int r = row + v;
    VRe[(m0 + r) * IMG_H + i0 + col]           = a00Re[v];
    VIm[(m0 + r) * IMG_H + i0 + col]           = a00Im[v];
    VRe[(m0 + r) * IMG_H + i0 + 16 + col]      = a01Re[v];
    VIm[(m0 + r) * IMG_H + i0 + 16 + col]      = a01Im[v];
    VRe[(m0 + 16 + r) * IMG_H + i0 + col]      = a10Re[v];
    VIm[(m0 + 16 + r) * IMG_H + i0 + col]      = a10Im[v];
    VRe[(m0 + 16 + r) * IMG_H + i0 + 16 + col] = a11Re[v];
    VIm[(m0 + 16 + r) * IMG_H + i0 + 16 + col] = a11Im[v];
  }
}

// ---------------------- forward stage 2 (row reduction) --------------------
// back[m] = sum_i Ax[m,i] * V[m,i].
// mode 0: Pipe update  w[m] /= |back[m]|
// mode 1: g[m] = back[m] * w[m]   (density-weighted k-space)
__global__ void __launch_bounds__(256) reduce_rows(
    const float* __restrict__ VRe, const float* __restrict__ VIm,
    const float* __restrict__ AxRe, const float* __restrict__ AxIm,
    float* __restrict__ w, float* __restrict__ gRe, float* __restrict__ gIm,
    int mode, int M) {
  const int lane = threadIdx.x & 31;
  const int wave = threadIdx.x >> 5;
  const int m = blockIdx.x * 8 + wave;
  if (m >= M) return;
  float sRe = 0.0f, sIm = 0.0f;
#pragma unroll
  for (int t = 0; t < 8; ++t) {
    const int i = lane + 32 * t;
    const float vr = VRe[m * IMG_H + i], vi = VIm[m * IMG_H + i];
    const float ar = AxRe[m * 256 + i], ai = AxIm[m * 256 + i];
    sRe += ar * vr - ai * vi;
    sIm += ar * vi + ai * vr;
  }
  sRe = wave_sum32(sRe);
  sIm = wave_sum32(sIm);
  if (lane == 0) {
    if (mode == 0) {
      const float mag = sqrtf(sRe * sRe + sIm * sIm);
      w[m] = w[m] / fmaxf(mag, 1e-20f);
    } else {
      const float wm = w[m];
      gRe[m] = sRe * wm;
      gIm[m] = sIm * wm;
    }
  }
}

// ----------------------------- normalization -------------------------------
__global__ void sumsq_kernel(const float* __restrict__ data, int n,
                             float* __restrict__ accum) {
  float s = 0.0f;
  for (int i = blockIdx.x * blockDim.x + threadIdx.x; i < n;
       i += gridDim.x * blockDim.x) {
    const float v = data[i];
    s += v * v;
  }
  s = wave_sum32(s);
  if ((threadIdx.x & 31) == 0) atomicAdd(accum, s);
}

__global__ void finalize_kernel(const float* __restrict__ outRe,
                                const float* __restrict__ outIm,
                                const float* __restrict__ accum,
                                float* __restrict__ dst, int n) {
  const int i = blockIdx.x * blockDim.x + threadIdx.x;
  if (i >= n) return;
  const float s = rsqrtf(fmaxf(*accum, 1e-30f));
  dst[2 * i + 0] = outRe[i] * s;  // complex64 interleaved
  dst[2 * i + 1] = outIm[i] * s;
}

// ------------------------------- launcher ----------------------------------
extern "C" void kernel_launch(void* const* d_in, const int* in_sizes, int n_in,
                              void* d_out, int out_size, void* d_ws,
                              size_t ws_size, hipStream_t stream) {
  const float* traj = (const float*)d_in[0];
  const float* x_re = (const float*)d_in[1];
  const float* x_im = (const float*)d_in[2];
  const int M = in_sizes[0] / 2;                 // 16384
  const int C = in_sizes[1] / (IMG_H * IMG_W);   // 4
  const size_t HW = (size_t)IMG_H * IMG_W;

  float* ws = (float*)d_ws;
  size_t off = 0;
  float* AxRe = ws + off; off += (size_t)M * 256;
  float* AxIm = ws + off; off += (size_t)M * 256;
  float* AyRe = ws + off; off += (size_t)M * 256;
  float* AyIm = ws + off; off += (size_t)M * 256;
  float* w    = ws + off; off += (size_t)M;
  float* imgRe = ws + off; off += HW;
  float* imgIm = ws + off; off += HW;
  float* VRe  = ws + off; off += (size_t)M * IMG_H;
  float* VIm  = ws + off; off += (size_t)M * IMG_H;
  float* gRe  = ws + off; off += (size_t)C * M;
  float* gIm  = ws + off; off += (size_t)C * M;
  float* outRe = ws + off; off += (size_t)C * HW;  // outIm must follow outRe
  float* outIm = ws + off; off += (size_t)C * HW;
  float* accum = ws + off; off += 1;

  build_tables<<<(M * 256 + 255) / 256, 256, 0, stream>>>(traj, AxRe, AxIm,
                                                          AyRe, AyIm, M);
  init_state<<<(M + 255) / 256, 256, 0, stream>>>(w, accum, M);

  const int gridBlocks   = (IMG_H / 16) * (IMG_W / 32);            // 128
  const int degridBlocks = ((M / 32) * (IMG_H / 32) + 7) / 8;      // 512

  // Pipe & Menon density compensation: w <- w / |G G^H w|
  for (int it = 0; it < PIPE_ITERS; ++it) {
    grid_adjoint_wmma<<<gridBlocks, 256, 0, stream>>>(
        w, nullptr, 0, AxRe, AxIm, AyRe, AyIm, imgRe, imgIm, M);
    degrid_stage1_wmma<<<degridBlocks, 256, 0, stream>>>(imgRe, imgIm, AyRe,
                                                         AyIm, VRe, VIm, M);
    reduce_rows<<<(M + 7) / 8, 256, 0, stream>>>(VRe, VIm, AxRe, AxIm, w,
                                                 nullptr, nullptr, 0, M);
  }

  // Per channel: forward NUDFT, density weight, adjoint NUDFT.
  for (int c = 0; c < C; ++c) {
    degrid_stage1_wmma<<<degridBlocks, 256, 0, stream>>>(
        x_re + (size_t)c * HW, x_im + (size_t)c * HW, AyRe, AyIm, VRe, VIm, M);
    reduce_rows<<<(M + 7) / 8, 256, 0, stream>>>(
        VRe, VIm, AxRe, AxIm, w, gRe + (size_t)c * M, gIm + (size_t)c * M, 1, M);
    grid_adjoint_wmma<<<gridBlocks, 256, 0, stream>>>(
        gRe + (size_t)c * M, gIm + (size_t)c * M, 1, AxRe, AxIm, AyRe, AyIm,
        outRe + (size_t)c * HW, outIm + (size_t)c * HW, M);
  }

  // Frobenius norm over the complex image, then scale + interleave to d_out.
  sumsq_kernel<<<512, 256, 0, stream>>>(outRe, (int)(2 * C * HW), accum);
  finalize_kernel<<<((int)(C * HW) + 255) / 256, 256, 0, stream>>>(
      outRe, outIm, accum, (float*)d_out, (int)(C * HW));
}